// MyModel_87522843561037
// MI455X (gfx1250) — compile-verified
//
#include <hip/hip_runtime.h>
#include <hip/hip_bf16.h>
#include <math.h>

// ---------------- problem constants (match reference) ----------------
constexpr int    N_        = 8000;
constexpr int    SEQ_      = 200;
constexpr int    MAX_STEPS = 199;

// ---------------- tuning ----------------
constexpr int    NPART     = 2048;            // reg-reduction blocks (partials)
constexpr int    RBLK      = 256;             // threads per reg block
constexpr int    TILE_F4   = RBLK * 4;        // float4 per LDS tile (16 KB)
// total float4 across pe+ne: 2*8000*8000/4 = 32,000,000 -> 31250 exact tiles
constexpr int    NUM_TILES = (2 * N_ * N_ / 4) / TILE_F4;

typedef __attribute__((ext_vector_type(2))) float v2f;
typedef __attribute__((ext_vector_type(8))) float v8f;
typedef int v4i_t __attribute__((vector_size(16)));   // matches builtin param type

// ---------------- CDNA5 feature probes ----------------
#if __has_builtin(__builtin_amdgcn_global_load_async_to_lds_b128)
#define HAVE_ASYNC 1
#else
#define HAVE_ASYNC 0
#endif

#if __has_builtin(__builtin_amdgcn_wmma_f32_16x16x4_f32)
#define HAVE_WMMA4 1
#else
#define HAVE_WMMA4 0
#endif

__device__ __forceinline__ void async_copy16(const float4* gsrc, float4* lds) {
#if HAVE_ASYNC
  __builtin_amdgcn_global_load_async_to_lds_b128(
      (__attribute__((address_space(1))) v4i_t*)(void*)gsrc,
      (__attribute__((address_space(3))) v4i_t*)(void*)lds,
      0, 0);
#else
  *lds = *gsrc;   // fallback: plain global load + ds_store
#endif
}

__device__ __forceinline__ void wait_async0() {
#if __has_builtin(__builtin_amdgcn_s_wait_asynccnt)
  __builtin_amdgcn_s_wait_asynccnt(0);
#else
  asm volatile("s_wait_asynccnt 0" ::: "memory");
#endif
}

// =====================================================================
// Kernel 1: loss from the sequential scan, collapsed to per-step gathers.
// Thread i replays the clip-chain of column a[i] over steps t < i.
// =====================================================================
__global__ __launch_bounds__(256)
void loss_kernel(const int* __restrict__ a, const float* __restrict__ s,
                 const float* __restrict__ pe, const float* __restrict__ ne,
                 const float* __restrict__ kp, float* __restrict__ ws) {
  __shared__ float ssh[SEQ_];
  __shared__ int   ash[SEQ_];
  __shared__ float red[256];

  const int tid = threadIdx.x;
  for (int j = tid; j < SEQ_; j += 256) { ssh[j] = s[j]; ash[j] = a[j]; }
  __syncthreads();

  float contrib = 0.0f;
  if (tid < MAX_STEPS) {
    const int ai = ash[tid];
    float kv = kp[ai];
    float v  = 1.0f;                       // running cumprod(s[0..t] >= 0)
    for (int t = 0; t < tid; ++t) {
      const float st = ssh[t];
      v *= (st >= 0.0f) ? 1.0f : 0.0f;     // valid[t] includes s[t]
      const size_t row = (size_t)ash[t] * (size_t)N_;
      const float d  = st * pe[row + ai] + (1.0f - st) * ne[row + ai];
      const float kn = fminf(fmaxf(kv + d, -30.0f), 30.0f);
      kv = (v > 0.0f) ? kn : kv;           // k = where(m>0, k_new, k)
    }
    v *= (ssh[tid] >= 0.0f) ? 1.0f : 0.0f; // m_i = valid[i]
    const float si = ssh[tid];
    const float p  = fminf(fmaxf(kv, 0.01f), 0.99f);
    const float li = -(si * logf(p) + (1.0f - si) * logf(1.0f - p));
    contrib = v * li;
  }

  red[tid] = contrib;
  __syncthreads();
  for (int off = 128; off > 0; off >>= 1) {
    if (tid < off) red[tid] += red[tid + off];
    __syncthreads();
  }
  if (tid == 0) ws[0] = red[0];
}

// =====================================================================
// Kernel 2: regularizer partials. 512 MB streamed through LDS with the
// CDNA5 async global->LDS copy path, double-buffered to overlap the HBM
// stream (~22us floor at 23.3 TB/s) with the v_log_f32 pipeline.
// =====================================================================
__global__ __launch_bounds__(RBLK)
void reg_kernel(const float* __restrict__ pe, const float* __restrict__ ne,
                float* __restrict__ ws) {
  __shared__ float4 buf[2][TILE_F4];   // 2 x 16 KB
  __shared__ float  red[RBLK];

  const int tid    = threadIdx.x;
  const int stride = gridDim.x;
  const int t0     = blockIdx.x;
  constexpr int HALF_F4 = N_ * N_ / 4;           // float4 count of one table

  const float4* peq = (const float4*)pe;
  const float4* neq = (const float4*)ne;

  auto issue = [&](int tile, int nb) {
    const size_t base = (size_t)tile * TILE_F4;
    const bool   in_pe = (base < (size_t)HALF_F4);          // tiles never straddle
    const float4* src  = in_pe ? (peq + base) : (neq + (base - HALF_F4));
#pragma unroll
    for (int j = 0; j < 4; ++j)
      async_copy16(src + tid + RBLK * j, &buf[nb][tid + RBLK * j]);
  };

  float acc = 0.0f;
  int nb = 0;
  if (t0 < NUM_TILES) issue(t0, 0);

  for (int t = t0; t < NUM_TILES; t += stride) {
    wait_async0();        // this wave's tile-t copies landed
    __syncthreads();      // everyone's copies landed; prior readers of buf[nb^1] done
    const int tn = t + stride;
    if (tn < NUM_TILES) issue(tn, nb ^ 1);
#pragma unroll
    for (int j = 0; j < 4; ++j) {
      const float4 x = buf[nb][tid + RBLK * j];
      acc += logf(fabsf(x.x) + 1.0f) + logf(fabsf(x.y) + 1.0f) +
             logf(fabsf(x.z) + 1.0f) + logf(fabsf(x.w) + 1.0f);
    }
    nb ^= 1;
  }

  red[tid] = acc;
  __syncthreads();
  for (int off = RBLK / 2; off > 0; off >>= 1) {
    if (tid < off) red[tid] += red[tid + off];
    __syncthreads();
  }
  if (tid == 0) ws[1 + blockIdx.x] = red[0];
}

// =====================================================================
// Kernel 3: single-wave finalize. Sums the 2048 partials with
// V_WMMA_F32_16X16X4_F32: A = ones(16x4) so D[m][n] = C[m][n] + colsum(B),
// accumulating through the f32 C operand (EXEC all-ones: one full wave,
// no divergence). Rows of D are identical -> lane-sum of d[0] = 2*total.
// =====================================================================
__global__ __launch_bounds__(32)
void finalize(const float* __restrict__ ws, float* __restrict__ out) {
  const int lane = threadIdx.x;   // one wave32, EXEC all 1s
  float total;

#if HAVE_WMMA4
  v8f acc = {};
  v2f ones; ones[0] = 1.0f; ones[1] = 1.0f;
  constexpr int ITERS = NPART / 64;     // 64 f32 values per WMMA (B is 4x16)
#pragma unroll 4
  for (int it = 0; it < ITERS; ++it) {
    const float* base = ws + 1 + it * 64;
    v2f b; b[0] = base[lane]; b[1] = base[32 + lane];
    // 8 args: (neg_a, A, neg_b, B, c_mod, C, reuse_a, reuse_b)
    acc = __builtin_amdgcn_wmma_f32_16x16x4_f32(
        false, ones, false, b, (short)0, acc, false, false);
  }
  float v = acc[0];                     // lanes 0..31 hold colsums (rows M=0, M=8)
  for (int m = 16; m >= 1; m >>= 1) v += __shfl_xor(v, m, 32);
  total = v * 0.5f;                     // two identical row copies across the wave
#else
  float v = 0.0f;
  for (int i = lane; i < NPART; i += 32) v += ws[1 + i];
  for (int m = 16; m >= 1; m >>= 1) v += __shfl_xor(v, m, 32);
  total = v;
#endif

  if (lane == 0) out[0] = ws[0] + 0.5f * total;
}

// =====================================================================
extern "C" void kernel_launch(void* const* d_in, const int* in_sizes, int n_in,
                              void* d_out, int out_size, void* d_ws, size_t ws_size,
                              hipStream_t stream) {
  const int*   a  = (const int*)d_in[0];
  const float* s  = (const float*)d_in[1];
  const float* pe = (const float*)d_in[2];
  const float* ne = (const float*)d_in[3];
  const float* kp = (const float*)d_in[4];
  float* out = (float*)d_out;
  float* ws  = (float*)d_ws;   // ws[0] = loss, ws[1..NPART] = reg partials

  reg_kernel<<<NPART, RBLK, 0, stream>>>(pe, ne, ws);
  loss_kernel<<<1, 256, 0, stream>>>(a, s, pe, ne, kp, ws);
  finalize<<<1, 32, 0, stream>>>(ws, out);
}